// RCNNTrainer_42494406427380
// MI455X (gfx1250) — compile-verified
//
#include <hip/hip_runtime.h>
#include <math.h>

// Problem constants (fixed by the reference).
#define B_   16
#define T_   64
#define R_   2048
#define C_   1001
#define N_   (B_*R_)   // 32768
#define K_   256
#define K3_  768
#define RED_ 16.0f

typedef __attribute__((ext_vector_type(2))) float v2f;
typedef __attribute__((ext_vector_type(8))) float v8f;
typedef unsigned int u32x4 __attribute__((ext_vector_type(4)));
typedef int          i32x4v __attribute__((ext_vector_type(4)));
typedef int          i32x8v __attribute__((ext_vector_type(8)));

// Exact 32-lane f32 sum using the matrix pipe:
// A (16x4) holds the 32 partials (one per lane in a[0], a[1]=0); B = ones.
// D[m,n] = rowsum(A,m), so summing the 8 D rows a lane-half holds and adding
// the other half via shfl_xor(16) yields the exact total in every lane.
// Layout-robust: any bijective A striping works since ALL rows get summed.
__device__ __forceinline__ float wave_sum_wmma(float p) {
  v2f a;  a[0] = p;    a[1] = 0.0f;
  v2f bm; bm[0] = 1.0f; bm[1] = 1.0f;
  v8f c = {};
  v8f d = __builtin_amdgcn_wmma_f32_16x16x4_f32(
      /*neg_a=*/false, a, /*neg_b=*/false, bm,
      /*c_mod=*/(short)0, c, /*reuse_a=*/false, /*reuse_b=*/false);
  float s = d[0] + d[1] + d[2] + d[3] + d[4] + d[5] + d[6] + d[7];
  s += __shfl_xor(s, 16, 32);
  return s;
}

// Tensor Data Mover: DMA one row of `nelts` f32 from global into LDS.
// D# group0: count=1 | lds_addr | global_addr | type=2 ("image").
// D# group1: workgroup_mask=0, data_size=2 (4B), tensor_dim0=tile_dim0=nelts,
//            tensor_dim1=tile_dim1=1, tensor_dim0_stride=nelts, no pad/iterate.
// Groups 2/3 unused (2-D tensor). Tracked with TENSORcnt.
__device__ __forceinline__ void tdm_load_row(const float* gsrc, unsigned lds_off,
                                             unsigned nelts) {
  const unsigned long long ga = (unsigned long long)(uintptr_t)gsrc;
  u32x4 g0;
  g0[0] = 1u;                                        // count=1 (user descriptor)
  g0[1] = lds_off;                                   // lds_addr (bytes)
  g0[2] = (unsigned)(ga & 0xffffffffu);              // global_addr[31:0]
  g0[3] = (unsigned)((ga >> 32) & 0x1ffffffu)        // global_addr[56:32]
        | (2u << 30);                                // type=2
  i32x8v g1;
  g1[0] = (int)(2u << 16);                           // data_size=4B, mask=0
  g1[1] = (int)(nelts << 16);                        // tensor_dim0[15:0] @ bit48
  g1[2] = (int)(1u << 16);                           // tensor_dim1=1 @ bit80
  g1[3] = (int)(nelts << 16);                        // tile_dim0 @ bit112
  g1[4] = 1;                                         // tile_dim1=1, tile_dim2=0
  g1[5] = (int)nelts;                                // tensor_dim0_stride[31:0]
  g1[6] = 0;
  g1[7] = 0;
  const i32x4v gz = {0, 0, 0, 0};
#if defined(__clang_major__) && (__clang_major__ >= 23)
  const i32x8v gz8 = {0, 0, 0, 0, 0, 0, 0, 0};
  __builtin_amdgcn_tensor_load_to_lds(g0, g1, gz, gz, gz8, 0);
#else
  __builtin_amdgcn_tensor_load_to_lds(g0, g1, gz, gz, 0);
#endif
}

// ---------------- Kernel 1: IoU max/argmax + fused gathers -----------------
// One thread per (b,r). GT boxes + classes of the image staged in LDS.
__global__ void k_iou(const float4* __restrict__ nms_reg4,
                      const float* __restrict__ bboxes,
                      const int*   __restrict__ classes,
                      int* __restrict__ mask, int* __restrict__ mcls,
                      float4* __restrict__ mbox4) {
  __shared__ float sbb[T_ * 4];
  __shared__ int   scls[T_];
  const int tid = threadIdx.x;
  const int g   = blockIdx.x * 256 + tid;   // flat (b,r); each block is one b
  const int b   = g >> 11;
  sbb[tid] = bboxes[b * T_ * 4 + tid];      // 256 floats == 64 boxes
  if (tid < T_) scls[tid] = classes[b * T_ + tid];
  __syncthreads();

  const float4 av = nms_reg4[g];            // global_load_b128
  const float a0 = av.x, a1 = av.y, a2 = av.z, a3 = av.w;
  const float area_a = (a2 - a0) * (a3 - a1);
  float best = -1.0f; int bi = 0;
  #pragma unroll 4
  for (int t = 0; t < T_; ++t) {
    const float b0 = sbb[t*4+0], b1 = sbb[t*4+1];
    const float b2 = sbb[t*4+2], b3 = sbb[t*4+3];
    const float ih = fmaxf(fminf(a2, b2) - fmaxf(a0, b0), 0.0f);
    const float iw = fmaxf(fminf(a3, b3) - fmaxf(a1, b1), 0.0f);
    const float inter  = ih * iw;
    const float area_b = (b2 - b0) * (b3 - b1);
    const float iou = inter / (area_a + area_b - inter + 1e-8f);
    if (iou > best) { best = iou; bi = t; }   // strict '>' == first argmax
  }
  mask[g] = (best > 0.5f) ? 1 : 0;
  mcls[g] = scls[bi];
  mbox4[g] = make_float4(sbb[bi*4+0], sbb[bi*4+1], sbb[bi*4+2], sbb[bi*4+3]);
}

// -------- Kernel 2: stable selection via deterministic prefix scan ---------
// Single block, 1024 threads x 32 elems. Matches stable argsort semantics:
// positives in ascending flat index (rank<K), negatives likewise (rank<K3).
// Padded slots stay 0 and are masked by `valid` downstream. No atomics ->
// bitwise-deterministic across graph replays.
__global__ void k_select(const int* __restrict__ mask,
                         int* __restrict__ pos_idx, int* __restrict__ neg_idx,
                         int* __restrict__ counts) {
  __shared__ int scan[1024];
  const int tid = threadIdx.x;
  if (tid < K_)  pos_idx[tid] = 0;
  if (tid < K3_) neg_idx[tid] = 0;
  const int i0 = tid * 32;
  int pc = 0;
  for (int k = 0; k < 32; ++k) pc += mask[i0 + k];
  scan[tid] = pc;
  __syncthreads();
  for (int off = 1; off < 1024; off <<= 1) {     // Hillis-Steele inclusive
    int v = scan[tid];
    int add = (tid >= off) ? scan[tid - off] : 0;
    __syncthreads();
    scan[tid] = v + add;
    __syncthreads();
  }
  int pr = scan[tid] - pc;                       // exclusive prefix of positives
  for (int k = 0; k < 32; ++k) {
    const int i = i0 + k;
    if (mask[i]) {
      if (pr < K_) pos_idx[pr] = i;
      ++pr;
    } else {
      const int nr = i - pr;                     // negatives before i
      if (nr < K3_) neg_idx[nr] = i;
    }
  }
  if (tid == 1023) counts[0] = scan[1023];       // total n_pos
}

// ------- Kernel 3: gathered log-softmax NLL + accuracy (1 wave / row) ------
// 8 waves/block. Each wave DMAs its 1001-float logits row into its private
// LDS slice via the Tensor Data Mover (1 instr instead of 32 loads+32 stores),
// waits on TENSORcnt, then runs max/argmax + sum-exp out of LDS; the 32-lane
// sum-exp reduction runs on the matrix pipe (v_wmma_f32_16x16x4_f32).
__global__ void k_cls(const float* __restrict__ rcnn_cls,
                      const int* __restrict__ pos_idx,
                      const int* __restrict__ neg_idx,
                      const int* __restrict__ counts,
                      const int* __restrict__ mcls,
                      float* __restrict__ nllv, float* __restrict__ accv) {
  __shared__ float srow[8][C_ + 7];
  const int tid  = threadIdx.x;
  const int lane = tid & 31;
  const int wv   = tid >> 5;
  const int j    = blockIdx.x * 8 + wv;          // 0..1023
  const int n_pos = counts[0];
  const int n_pos_eff = min(n_pos, K_);
  const int n_neg_eff = min(N_ - n_pos, K3_);

  int src, tgt; float valid;                     // wave-uniform branch (SALU)
  if (j < K_) {
    src = pos_idx[j];
    tgt = mcls[src];
    valid = (j < n_pos_eff) ? 1.0f : 0.0f;
  } else {
    src = neg_idx[j - K_];
    tgt = 0;
    valid = ((j - K_) < n_neg_eff) ? 1.0f : 0.0f;
  }

  // TDM: stage this wave's gathered row into its LDS slice.
  float* myrow = &srow[wv][0];
  const unsigned lds_off = (unsigned)(size_t)(void*)myrow;  // LDS byte offset
  tdm_load_row(rcnn_cls + (size_t)src * C_, lds_off, C_);
  __builtin_amdgcn_s_wait_tensorcnt(0);
  asm volatile("" : : "r"(myrow) : "memory");    // TDM wrote LDS: no hoisting
  __syncthreads();                               // uniform; compiler mem fence

  float bv = -3.4e38f; int bi = 0x7fffffff;
  for (int i = lane; i < C_; i += 32) {
    const float x = myrow[i];
    if (x > bv) { bv = x; bi = i; }              // keeps earliest per lane
  }
  for (int off = 16; off >= 1; off >>= 1) {      // max, tie -> smaller index
    const float ov = __shfl_xor(bv, off, 32);
    const int   oi = __shfl_xor(bi, off, 32);
    if (ov > bv || (ov == bv && oi < bi)) { bv = ov; bi = oi; }
  }
  const float m = bv;
  float s = 0.0f;
  for (int i = lane; i < C_; i += 32) s += expf(myrow[i] - m);
  const float stot = wave_sum_wmma(s);           // exact 32-lane sum (matrix pipe)
  const float xt   = myrow[tgt];
  const float nll  = m + logf(stot) - xt;
  if (lane == 0) {
    nllv[j] = valid * nll;
    accv[j] = valid * ((bi == tgt) ? 1.0f : 0.0f);
  }
}

// ---- Kernel 4: smooth-L1 regression + final deterministic reductions ------
__global__ void k_final(const float* __restrict__ nllv,
                        const float* __restrict__ accv,
                        const int* __restrict__ pos_idx,
                        const int* __restrict__ counts,
                        const float4* __restrict__ rcnn_reg4,
                        const float4* __restrict__ nms_reg4,
                        const float4* __restrict__ mbox4,
                        float* __restrict__ out) {
  __shared__ float part[8][6];
  const int tid  = threadIdx.x;                  // 256 threads, 8 waves
  const int lane = tid & 31;
  const int wv   = tid >> 5;
  const int n_pos = counts[0];
  const int n_pos_eff = min(n_pos, K_);
  const int n_neg_eff = min(N_ - n_pos, K3_);

  const float p_nll = nllv[tid] + nllv[tid+256] + nllv[tid+512] + nllv[tid+768];
  const float p_acc = accv[tid] + accv[tid+256] + accv[tid+512] + accv[tid+768];

  // Regression for positive slot `tid` (branch-free; masked by w).
  const int    src = pos_idx[tid];
  const float  w   = (tid < n_pos_eff) ? 1.0f : 0.0f;
  const float4 mn  = nms_reg4[src];
  const float4 mr  = rcnn_reg4[src];
  const float4 mb  = mbox4[src];
  const float rd0 = floorf(mn.x * RED_) / RED_;
  const float rd1 = floorf(mn.y * RED_) / RED_;
  const float rd2 = ceilf (mn.z * RED_) / RED_;
  const float rd3 = ceilf (mn.w * RED_) / RED_;
  auto sl1 = [](float pred, float t) -> float {
    const float d = pred - t, ad = fabsf(d);
    return (ad < 1.0f) ? 0.5f * d * d : ad - 0.5f;
  };
  const float r0 = w * sl1(mr.x, mb.x - rd0);
  const float r1 = w * sl1(mr.y, mb.y - rd1);
  const float r2 = w * sl1(mr.z, mb.z - rd2);
  const float r3 = w * sl1(mr.w, mb.w - rd3);

  // Six exact 32-lane sums on the matrix pipe (uniform call sites, EXEC=all-1).
  const float s_nll = wave_sum_wmma(p_nll);
  const float s_acc = wave_sum_wmma(p_acc);
  const float s_r0  = wave_sum_wmma(r0);
  const float s_r1  = wave_sum_wmma(r1);
  const float s_r2  = wave_sum_wmma(r2);
  const float s_r3  = wave_sum_wmma(r3);
  if (lane == 0) {
    part[wv][0] = s_nll; part[wv][1] = s_acc;
    part[wv][2] = s_r0;  part[wv][3] = s_r1;
    part[wv][4] = s_r2;  part[wv][5] = s_r3;
  }
  __syncthreads();
  if (tid == 0) {
    float tn = 0.f, ta = 0.f, t0 = 0.f, t1 = 0.f, t2 = 0.f, t3 = 0.f;
    for (int q = 0; q < 8; ++q) {
      tn += part[q][0]; ta += part[q][1];
      t0 += part[q][2]; t1 += part[q][3];
      t2 += part[q][4]; t3 += part[q][5];
    }
    const float denom = fmaxf((float)(n_pos_eff + n_neg_eff), 1.0f);
    const float rden  = fmaxf((float)n_pos_eff, 1.0f);
    const float reg   = (t0 + t1 + t2 + t3) / rden;
    out[0] = tn / denom;                               // cls_loss
    out[1] = ((n_pos > 0) ? reg : 0.0f) * 0.25f;       // reg_loss / 4
    out[2] = ta / denom;                               // accuracy
  }
}

extern "C" void kernel_launch(void* const* d_in, const int* in_sizes, int n_in,
                              void* d_out, int out_size, void* d_ws, size_t ws_size,
                              hipStream_t stream) {
  const float* nms_reg  = (const float*)d_in[0];
  // d_in[1] = nms_cls: unused by the reference.
  const float* rcnn_reg = (const float*)d_in[2];
  const float* rcnn_cls = (const float*)d_in[3];
  const float* bboxes   = (const float*)d_in[4];
  const int*   classes  = (const int*)d_in[5];

  char* ws = (char*)d_ws;
  int*   mask    = (int*)ws;    ws += (size_t)N_ * 4;
  int*   mcls    = (int*)ws;    ws += (size_t)N_ * 4;
  float* mbox    = (float*)ws;  ws += (size_t)N_ * 16;
  int*   pos_idx = (int*)ws;    ws += K_ * 4;
  int*   neg_idx = (int*)ws;    ws += K3_ * 4;
  int*   counts  = (int*)ws;    ws += 16;
  float* nllv    = (float*)ws;  ws += 1024 * 4;
  float* accv    = (float*)ws;

  k_iou   <<<N_ / 256, 256, 0, stream>>>((const float4*)nms_reg, bboxes, classes,
                                         mask, mcls, (float4*)mbox);
  k_select<<<1, 1024, 0, stream>>>(mask, pos_idx, neg_idx, counts);
  k_cls   <<<128, 256, 0, stream>>>(rcnn_cls, pos_idx, neg_idx, counts, mcls, nllv, accv);
  k_final <<<1, 256, 0, stream>>>(nllv, accv, pos_idx, counts,
                                  (const float4*)rcnn_reg, (const float4*)nms_reg,
                                  (const float4*)mbox, (float*)d_out);
}